// SpatialEmbeddingCrossAttentionBlock_17789754540913
// MI455X (gfx1250) — compile-verified
//
#include <hip/hip_runtime.h>
#include <hip/hip_bf16.h>

// ---------------- CDNA5 WMMA f32 16x16x4 helpers ----------------
typedef float v2f __attribute__((ext_vector_type(2)));
typedef float v8f __attribute__((ext_vector_type(8)));

static __device__ __forceinline__ v8f wmma4(v2f a, v2f b, v8f c) {
  // 8 args: (neg_a, A, neg_b, B, c_mod, C, reuse_a, reuse_b)
  return __builtin_amdgcn_wmma_f32_16x16x4_f32(false, a, false, b, (short)0, c,
                                               false, false);
}

#define IMIN(a, b) ((a) < (b) ? (a) : (b))

// problem constants
#define B_    8
#define IN_   320
#define HW_   24
#define L_    576
#define CTX_  768
#define DTOK_ 64
#define HEADS_ 8
#define HC_   64
#define INNER_ 512
#define FF_   2048
#define BH_   64
#define G_    32

enum { A_ROW = 0, A_COL = 1 };
enum { B_ROW = 0, B_TRN = 1 };
enum { E_NONE = 0, E_BIAS = 1, E_GELU = 2, E_RES = 4, E_SCALE = 8 };

// ---------------------------------------------------------------------------
// Generic batched WMMA GEMM: C[z] = op(A[z]) * op(B[z])  (+ epilogue)
//   A_ROW: A[m][k] = A[m*lda + k]      A_COL: A[m][k] = A[k*lda + m]
//   B_ROW: B[k][n] = B[k*ldb + n]      B_TRN: B[k][n] = B[n*ldb + k]
// Block = 256 threads = 8 wave32s arranged 2(M) x 4(N); wave tile 32x32
// (2x2 WMMA fragments). Block tile 64(M) x 128(N). K multiple of 4.
// EXEC stays full through all WMMAs (loads use clamped indices; only the
// stores are guarded), per the CDNA5 WMMA EXEC restriction.
// ---------------------------------------------------------------------------
template <int AM, int BM, int EPI>
__global__ __launch_bounds__(256) void wmma_gemm_kernel(
    const float* __restrict__ Ap, const float* __restrict__ Bp,
    float* __restrict__ Cp, const float* __restrict__ bias,
    const float* __restrict__ resid, int M, int N, int K, int lda, int ldb,
    int ldc, long long sA, long long sB, long long sC, long long sRes,
    float scale) {
  const int z = blockIdx.z;
  const float* A = Ap + (size_t)z * (size_t)sA;
  const float* B = Bp + (size_t)z * (size_t)sB;
  float* C = Cp + (size_t)z * (size_t)sC;
  const float* R = resid ? (resid + (size_t)z * (size_t)sRes) : nullptr;

  const int lane = threadIdx.x & 31;
  const int wave = threadIdx.x >> 5;
  const int ln = lane & 15;   // M (for A) / N (for B,C) lane index
  const int hi = lane >> 4;   // half-wave selects K pair / C row block
  const int wm = wave >> 2;   // 0..1
  const int wn = wave & 3;    // 0..3
  const int m0 = blockIdx.y * 64 + wm * 32;
  const int n0 = blockIdx.x * 128 + wn * 32;

  const v8f zero8 = {0.f, 0.f, 0.f, 0.f, 0.f, 0.f, 0.f, 0.f};
  v8f acc[2][2];
  acc[0][0] = zero8; acc[0][1] = zero8; acc[1][0] = zero8; acc[1][1] = zero8;

  // clamped load indices (keep EXEC full; garbage rows are never stored)
  const int ar0 = IMIN(m0 + ln, M - 1);
  const int ar1 = IMIN(m0 + 16 + ln, M - 1);
  const int bc0 = IMIN(n0 + ln, N - 1);
  const int bc1 = IMIN(n0 + 16 + ln, N - 1);

  for (int k0 = 0; k0 < K; k0 += 4) {
    const int kk = k0 + 2 * hi;
    v2f a0, a1, b0, b1;
    if (AM == A_ROW) {
      a0 = *(const v2f*)(A + (size_t)ar0 * lda + kk);
      a1 = *(const v2f*)(A + (size_t)ar1 * lda + kk);
    } else {
      const float* p0 = A + (size_t)kk * lda;
      const float* p1 = A + (size_t)(kk + 1) * lda;
      a0.x = p0[ar0]; a0.y = p1[ar0];
      a1.x = p0[ar1]; a1.y = p1[ar1];
    }
    if (BM == B_ROW) {
      const float* p0 = B + (size_t)kk * ldb;
      const float* p1 = B + (size_t)(kk + 1) * ldb;
      b0.x = p0[bc0]; b0.y = p1[bc0];
      b1.x = p0[bc1]; b1.y = p1[bc1];
    } else {
      b0 = *(const v2f*)(B + (size_t)bc0 * ldb + kk);
      b1 = *(const v2f*)(B + (size_t)bc1 * ldb + kk);
    }
    acc[0][0] = wmma4(a0, b0, acc[0][0]);
    acc[0][1] = wmma4(a0, b1, acc[0][1]);
    acc[1][0] = wmma4(a1, b0, acc[1][0]);
    acc[1][1] = wmma4(a1, b1, acc[1][1]);
  }

#pragma unroll
  for (int i = 0; i < 2; ++i) {
#pragma unroll
    for (int j = 0; j < 2; ++j) {
#pragma unroll
      for (int r = 0; r < 8; ++r) {
        const int row = m0 + i * 16 + r + 8 * hi;  // C/D: VGPR r <-> M=r+8*hi
        const int col = n0 + j * 16 + ln;          //      lane  <-> N
        if (row < M && col < N) {
          float v = acc[i][j][r];
          if (EPI & E_SCALE) v *= scale;
          if (EPI & E_BIAS) v += bias[row];
          if (EPI & E_GELU)
            v = 0.5f * v * (1.0f + erff(v * 0.70710678118654752f));
          if (EPI & E_RES) v += R[(size_t)row * ldc + col];
          C[(size_t)row * ldc + col] = v;
        }
      }
    }
  }
}

// ---------------------------------------------------------------------------
// Context projection GEMV: y[b,o] = sum_k W[o,k]*c[b,k] + bias[o]
// grid (M/256, B), c row cached in LDS.
// ---------------------------------------------------------------------------
__global__ __launch_bounds__(256) void cproj_kernel(
    const float* __restrict__ W, const float* __restrict__ bias,
    const float* __restrict__ c, float* __restrict__ y) {
  __shared__ float cs[CTX_];
  const int b = blockIdx.y;
  for (int i = threadIdx.x; i < CTX_; i += 256) cs[i] = c[(size_t)b * CTX_ + i];
  __syncthreads();
  const int o = blockIdx.x * 256 + threadIdx.x;
  const float* w = W + (size_t)o * CTX_;
  float acc = 0.f;
  for (int k = 0; k < CTX_; ++k) acc += w[k] * cs[k];
  y[(size_t)b * (CTX_ * DTOK_) + o] = acc + bias[o];
}

// ---------------------------------------------------------------------------
// GroupNorm over [Bn, C, S] with G groups (biased var, eps=1e-5).
// One block per (b, group).
// ---------------------------------------------------------------------------
__global__ __launch_bounds__(256) void groupnorm_kernel(
    const float* __restrict__ x, float* __restrict__ y,
    const float* __restrict__ g, const float* __restrict__ bt, int C, int G,
    int S, float eps) {
  const int b = blockIdx.x / G;
  const int gr = blockIdx.x % G;
  const int Cg = C / G;
  const int count = Cg * S;
  const float* xb = x + ((size_t)b * C + (size_t)gr * Cg) * S;
  float* yb = y + ((size_t)b * C + (size_t)gr * Cg) * S;

  float s = 0.f, s2 = 0.f;
  for (int i = threadIdx.x; i < count; i += 256) {
    float v = xb[i];
    s += v;
    s2 += v * v;
  }
  __shared__ float r1[256], r2[256];
  r1[threadIdx.x] = s;
  r2[threadIdx.x] = s2;
  __syncthreads();
  for (int off = 128; off > 0; off >>= 1) {
    if (threadIdx.x < off) {
      r1[threadIdx.x] += r1[threadIdx.x + off];
      r2[threadIdx.x] += r2[threadIdx.x + off];
    }
    __syncthreads();
  }
  const float inv = 1.0f / (float)count;
  const float mean = r1[0] * inv;
  const float var = r2[0] * inv - mean * mean;
  const float rs = rsqrtf(var + eps);
  for (int i = threadIdx.x; i < count; i += 256) {
    const int ch = gr * Cg + i / S;
    yb[i] = (xb[i] - mean) * rs * g[ch] + bt[ch];
  }
}

// ---------------------------------------------------------------------------
// Row softmax, row length 576, one block per row.
// ---------------------------------------------------------------------------
__global__ __launch_bounds__(256) void softmax576_kernel(
    const float* __restrict__ in, float* __restrict__ out) {
  __shared__ float red[256];
  const size_t row = blockIdx.x;
  const float* r = in + row * L_;
  float* o = out + row * L_;
  const int tid = threadIdx.x;

  float mx = -3.402823466e38f;
  for (int i = tid; i < L_; i += 256) mx = fmaxf(mx, r[i]);
  red[tid] = mx;
  __syncthreads();
  for (int off = 128; off > 0; off >>= 1) {
    if (tid < off) red[tid] = fmaxf(red[tid], red[tid + off]);
    __syncthreads();
  }
  mx = red[0];
  __syncthreads();

  float s = 0.f;
  for (int i = tid; i < L_; i += 256) s += expf(r[i] - mx);
  red[tid] = s;
  __syncthreads();
  for (int off = 128; off > 0; off >>= 1) {
    if (tid < off) red[tid] += red[tid + off];
    __syncthreads();
  }
  const float inv = 1.0f / red[0];
  for (int i = tid; i < L_; i += 256) o[i] = expf(r[i] - mx) * inv;
}

// ---------------------------------------------------------------------------
// Cross-attn: blend raw row stats (ddof=1) with src_map row stats, then
// softmax over Lk=64. One block of 64 threads per (bh, t) row.
// ---------------------------------------------------------------------------
__global__ __launch_bounds__(64) void cross_blend_softmax_kernel(
    const float* __restrict__ raw, const float* __restrict__ src,
    float* __restrict__ out) {
  __shared__ float red[64];
  const size_t row = blockIdx.x;
  const int t = threadIdx.x;
  const float x = raw[row * DTOK_ + t];
  const float sv = src[row * DTOK_ + t];

#define RED_SUM()                                        \
  __syncthreads();                                       \
  for (int off = 32; off > 0; off >>= 1) {               \
    if (t < off) red[t] += red[t + off];                 \
    __syncthreads();                                     \
  }
#define RED_MAX()                                        \
  __syncthreads();                                       \
  for (int off = 32; off > 0; off >>= 1) {               \
    if (t < off) red[t] = fmaxf(red[t], red[t + off]);   \
    __syncthreads();                                     \
  }

  red[t] = x;
  RED_SUM();
  const float mean = red[0] * (1.0f / 64.0f);
  __syncthreads();
  float d = x - mean;
  red[t] = d * d;
  RED_SUM();
  const float stdv = sqrtf(red[0] * (1.0f / 63.0f));  // ddof=1
  __syncthreads();

  red[t] = sv;
  RED_SUM();
  const float mean_s = red[0] * (1.0f / 64.0f);
  __syncthreads();
  d = sv - mean_s;
  red[t] = d * d;
  RED_SUM();
  const float std_s = sqrtf(red[0] * (1.0f / 63.0f));
  __syncthreads();

  const float tmean = 0.5f * (mean_s + mean);
  const float tstd = 0.5f * (std_s + stdv);
  const float a = (x - mean) / stdv * tstd + tmean;

  red[t] = a;
  RED_MAX();
  const float mx = red[0];
  __syncthreads();
  const float e = expf(a - mx);
  red[t] = e;
  RED_SUM();
  out[row * DTOK_ + t] = e / red[0];
#undef RED_SUM
#undef RED_MAX
}

// ---------------------------------------------------------------------------
extern "C" void kernel_launch(void* const* d_in, const int* in_sizes, int n_in,
                              void* d_out, int out_size, void* d_ws,
                              size_t ws_size, hipStream_t stream) {
  const float* x       = (const float*)d_in[0];
  const float* c       = (const float*)d_in[1];
  const float* src_map = (const float*)d_in[2];
  const float* w_xin   = (const float*)d_in[3];
  const float* b_xin   = (const float*)d_in[4];
  const float* w_cproj = (const float*)d_in[5];
  const float* b_cproj = (const float*)d_in[6];
  const float* g_cn    = (const float*)d_in[7];
  const float* bt_cn   = (const float*)d_in[8];
  const float* g_xn    = (const float*)d_in[9];
  const float* bt_xn   = (const float*)d_in[10];
  const float* wq_s = (const float*)d_in[11]; const float* bq_s = (const float*)d_in[12];
  const float* wk_s = (const float*)d_in[13]; const float* bk_s = (const float*)d_in[14];
  const float* wv_s = (const float*)d_in[15]; const float* bv_s = (const float*)d_in[16];
  const float* wo_s = (const float*)d_in[17]; const float* bo_s = (const float*)d_in[18];
  const float* wq_c = (const float*)d_in[19]; const float* bq_c = (const float*)d_in[20];
  const float* wk_c = (const float*)d_in[21]; const float* bk_c = (const float*)d_in[22];
  const float* wv_c = (const float*)d_in[23]; const float* bv_c = (const float*)d_in[24];
  const float* wo_c = (const float*)d_in[25]; const float* bo_c = (const float*)d_in[26];
  const float* w_ff1 = (const float*)d_in[27]; const float* b_ff1 = (const float*)d_in[28];
  const float* w_ff2 = (const float*)d_in[29]; const float* b_ff2 = (const float*)d_in[30];
  const float* w_xout = (const float*)d_in[31]; const float* b_xout = (const float*)d_in[32];

  // output regions (concatenated in return order)
  float* out = (float*)d_out;                                  // [8,320,24,24]
  float* self_map = out + (size_t)B_ * IN_ * L_;               // [64,576,576]
  float* cross_map = self_map + (size_t)BH_ * L_ * L_;         // [64,576,64]

  // per-batch strides (floats)
  const long long sX  = (long long)IN_ * L_;      // 184320
  const long long sI  = (long long)INNER_ * L_;   // 294912
  const long long sFF = (long long)FF_ * L_;      // 1179648
  const long long sHL = (long long)HC_ * L_;      // 36864 (per-head q/k/v slice)
  const long long sLL = (long long)L_ * L_;       // 331776
  const long long sCF = (long long)CTX_ * DTOK_;  // 49152
  const long long sKC = (long long)INNER_ * DTOK_;// 32768
  const long long sHD = (long long)HC_ * DTOK_;   // 4096
  const long long sCM = (long long)L_ * DTOK_;    // 36864

  // workspace layout (floats)
  float* ws  = (float*)d_ws;
  float* xf  = ws;                    // residual stream [B,512,576]
  float* xn  = xf + (size_t)B_ * sI;  // normalized / attn-out temp
  float* qb  = xn + (size_t)B_ * sI;
  float* kb  = qb + (size_t)B_ * sI;
  float* vb  = kb + (size_t)B_ * sI;
  float* cfb = vb + (size_t)B_ * sI;         // [B,768,64]
  float* big = cfb + (size_t)B_ * sCF;       // attnw (85MB) / crossw / h1

  const dim3 blk(256);

  // 1) xf = w_xin * x + b_xin            (M=512,N=576,K=320)
  wmma_gemm_kernel<A_ROW, B_ROW, E_BIAS><<<dim3(5, 8, B_), blk, 0, stream>>>(
      w_xin, x, xf, b_xin, nullptr, INNER_, L_, IN_, IN_, L_, L_, 0, sX, sI, 0,
      1.0f);

  // 2) cf = c @ w_cproj^T + b ; 3) GroupNorm(cf) in place
  cproj_kernel<<<dim3((CTX_ * DTOK_) / 256, B_), blk, 0, stream>>>(
      w_cproj, b_cproj, c, cfb);
  groupnorm_kernel<<<dim3(B_ * G_), blk, 0, stream>>>(cfb, cfb, g_cn, bt_cn,
                                                      CTX_, G_, DTOK_, 1e-5f);

  // ---------------- self attention ----------------
  groupnorm_kernel<<<dim3(B_ * G_), blk, 0, stream>>>(xf, xn, g_xn, bt_xn,
                                                      INNER_, G_, L_, 1e-5f);
  wmma_gemm_kernel<A_ROW, B_ROW, E_BIAS><<<dim3(5, 8, B_), blk, 0, stream>>>(
      wq_s, xn, qb, bq_s, nullptr, INNER_, L_, INNER_, INNER_, L_, L_, 0, sI,
      sI, 0, 1.0f);
  wmma_gemm_kernel<A_ROW, B_ROW, E_BIAS><<<dim3(5, 8, B_), blk, 0, stream>>>(
      wk_s, xn, kb, bk_s, nullptr, INNER_, L_, INNER_, INNER_, L_, L_, 0, sI,
      sI, 0, 1.0f);
  wmma_gemm_kernel<A_ROW, B_ROW, E_BIAS><<<dim3(5, 8, B_), blk, 0, stream>>>(
      wv_s, xn, vb, bv_s, nullptr, INNER_, L_, INNER_, INNER_, L_, L_, 0, sI,
      sI, 0, 1.0f);
  // raw[t,s] = 0.125 * q^T k   (per bh; M=576,N=576,K=64) -> self_map output
  wmma_gemm_kernel<A_COL, B_ROW, E_SCALE><<<dim3(5, 9, BH_), blk, 0, stream>>>(
      qb, kb, self_map, nullptr, nullptr, L_, L_, HC_, L_, L_, L_, sHL, sHL,
      sLL, 0, 0.125f);
  softmax576_kernel<<<dim3(BH_ * L_), blk, 0, stream>>>(self_map, big);
  // a[c,t] = v * w^T  (M=64,N=576,K=576) -> xn reused as attn-out
  wmma_gemm_kernel<A_ROW, B_TRN, E_NONE><<<dim3(5, 1, BH_), blk, 0, stream>>>(
      vb, big, xn, nullptr, nullptr, HC_, L_, L_, L_, L_, L_, sHL, sLL, sHL, 0,
      1.0f);
  // xf += wo_s * a + bo_s
  wmma_gemm_kernel<A_ROW, B_ROW, E_BIAS | E_RES>
      <<<dim3(5, 8, B_), blk, 0, stream>>>(wo_s, xn, xf, bo_s, xf, INNER_, L_,
                                           INNER_, INNER_, L_, L_, 0, sI, sI,
                                           sI, 1.0f);

  // ---------------- cross attention ----------------
  groupnorm_kernel<<<dim3(B_ * G_), blk, 0, stream>>>(xf, xn, g_xn, bt_xn,
                                                      INNER_, G_, L_, 1e-5f);
  wmma_gemm_kernel<A_ROW, B_ROW, E_BIAS><<<dim3(5, 8, B_), blk, 0, stream>>>(
      wq_c, xn, qb, bq_c, nullptr, INNER_, L_, INNER_, INNER_, L_, L_, 0, sI,
      sI, 0, 1.0f);
  // k,v from context: M=512,N=64,K=768
  wmma_gemm_kernel<A_ROW, B_ROW, E_BIAS><<<dim3(1, 8, B_), blk, 0, stream>>>(
      wk_c, cfb, kb, bk_c, nullptr, INNER_, DTOK_, CTX_, CTX_, DTOK_, DTOK_, 0,
      sCF, sKC, 0, 1.0f);
  wmma_gemm_kernel<A_ROW, B_ROW, E_BIAS><<<dim3(1, 8, B_), blk, 0, stream>>>(
      wv_c, cfb, vb, bv_c, nullptr, INNER_, DTOK_, CTX_, CTX_, DTOK_, DTOK_, 0,
      sCF, sKC, 0, 1.0f);
  // raw cross scores (M=576,N=64,K=64) -> cross_map output
  wmma_gemm_kernel<A_COL, B_ROW, E_SCALE><<<dim3(1, 9, BH_), blk, 0, stream>>>(
      qb, kb, cross_map, nullptr, nullptr, L_, DTOK_, HC_, L_, DTOK_, DTOK_,
      sHL, sHD, sCM, 0, 0.125f);
  cross_blend_softmax_kernel<<<dim3(BH_ * L_), dim3(64), 0, stream>>>(
      cross_map, src_map, big);
  // a[c,t] (M=64,N=576,K=64)
  wmma_gemm_kernel<A_ROW, B_TRN, E_NONE><<<dim3(5, 1, BH_), blk, 0, stream>>>(
      vb, big, xn, nullptr, nullptr, HC_, L_, DTOK_, DTOK_, DTOK_, L_, sHD,
      sCM, sHL, 0, 1.0f);
  wmma_gemm_kernel<A_ROW, B_ROW, E_BIAS | E_RES>
      <<<dim3(5, 8, B_), blk, 0, stream>>>(wo_c, xn, xf, bo_c, xf, INNER_, L_,
                                           INNER_, INNER_, L_, L_, 0, sI, sI,
                                           sI, 1.0f);

  // ---------------- FFN ----------------
  groupnorm_kernel<<<dim3(B_ * G_), blk, 0, stream>>>(xf, xn, g_xn, bt_xn,
                                                      INNER_, G_, L_, 1e-5f);
  // h1[f,l] = gelu(w_ff1 * xn + b_ff1)   (M=2048,N=576,K=512)
  wmma_gemm_kernel<A_ROW, B_ROW, E_BIAS | E_GELU>
      <<<dim3(5, 32, B_), blk, 0, stream>>>(w_ff1, xn, big, b_ff1, nullptr,
                                            FF_, L_, INNER_, INNER_, L_, L_, 0,
                                            sI, sFF, 0, 1.0f);
  // xf += w_ff2 * h1 + b_ff2             (M=512,N=576,K=2048)
  wmma_gemm_kernel<A_ROW, B_ROW, E_BIAS | E_RES>
      <<<dim3(5, 8, B_), blk, 0, stream>>>(w_ff2, big, xf, b_ff2, xf, INNER_,
                                           L_, FF_, FF_, L_, L_, 0, sFF, sI,
                                           sI, 1.0f);

  // ---------------- output projection + input residual ----------------
  wmma_gemm_kernel<A_ROW, B_ROW, E_BIAS | E_RES>
      <<<dim3(5, 5, B_), blk, 0, stream>>>(w_xout, xf, out, b_xout, x, IN_, L_,
                                           INNER_, INNER_, L_, L_, 0, sI, sX,
                                           sX, 1.0f);
}